// GNN_5394478924400
// MI455X (gfx1250) — compile-verified
//
#include <hip/hip_runtime.h>

// CDNA5 GCN (2-layer) for MI455X / gfx1250 — wave32, f32 WMMA 16x16x4.
// Pipeline: deg -> dinv -> [GEMM(WMMA) -> init(self-loop+bias) -> edge scatter (f32 atomics)] x2, ReLU between.

typedef __attribute__((ext_vector_type(2))) float v2f;
typedef __attribute__((ext_vector_type(8))) float v8f;

#define IN_DIM 128

// ---------------- degree / norm ----------------

__global__ void fill_deg_kernel(float* __restrict__ deg, int n) {
  int i = blockIdx.x * blockDim.x + threadIdx.x;
  if (i < n) deg[i] = 1.0f;  // self-loop contributes 1
}

__global__ void deg_accum_kernel(const int* __restrict__ tgt, float* __restrict__ deg, int e) {
  int i = blockIdx.x * blockDim.x + threadIdx.x;
  if (i < e) atomicAdd(&deg[tgt[i]], 1.0f);
}

__global__ void rsqrt_kernel(float* __restrict__ d, int n) {
  int i = blockIdx.x * blockDim.x + threadIdx.x;
  if (i < n) d[i] = rsqrtf(d[i]);  // deg >= 1 always (self-loops)
}

// ---------------- WMMA GEMM: H[n_rows x NC] = X[n_rows x 128] @ W[128 x NC] ----------------
// Per-wave 16-row strip, NC/16 column tiles, K in steps of 4 via V_WMMA_F32_16X16X4_F32.
// A layout: lane<16 -> row=lane, K={0,1}; lane>=16 -> K={2,3}.
// B layout: lane<16 -> col=lane, K={0,1}; lane>=16 -> K={2,3}. (W^T staged in LDS -> contiguous b64)
// D layout: col = lane&15, rows j + 8*(lane>>4), j=0..7.

template <int NC>
__global__ __launch_bounds__(256) void gemm_wmma_f32(const float* __restrict__ X,
                                                     const float* __restrict__ W,
                                                     float* __restrict__ H, int n_rows) {
  __shared__ __align__(16) float Wt[NC * IN_DIM];  // Wt[c*128 + k] = W[k*NC + c]
  for (int i = threadIdx.x; i < NC * IN_DIM; i += 256) {
    int k = i / NC;
    int c = i - k * NC;
    Wt[c * IN_DIM + k] = W[i];
  }
  __syncthreads();

  const int lane = threadIdx.x & 31;
  const int wave = threadIdx.x >> 5;
  const int row_base = blockIdx.x * 128 + wave * 16;
  if (row_base + 16 > n_rows) return;  // N is a multiple of 16 -> tiles are exact

  const int r = lane & 15;             // A-row / B-col / D-col within tile
  const int koff = (lane >> 4) << 1;   // lane-half K sub-offset {0,2}

  const float* xrow = X + (size_t)(row_base + r) * IN_DIM + koff;

  constexpr int NT = NC / 16;
  v8f acc[NT];
  const v8f vzero = {0.f, 0.f, 0.f, 0.f, 0.f, 0.f, 0.f, 0.f};
#pragma unroll
  for (int t = 0; t < NT; ++t) acc[t] = vzero;

#pragma unroll 4
  for (int k0 = 0; k0 < IN_DIM; k0 += 4) {
    v2f a = *(const v2f*)(xrow + k0);
#pragma unroll
    for (int t = 0; t < NT; ++t) {
      v2f b = *(const v2f*)(&Wt[(t * 16 + r) * IN_DIM + k0 + koff]);
      acc[t] = __builtin_amdgcn_wmma_f32_16x16x4_f32(false, a, false, b,
                                                     (short)0, acc[t], false, false);
    }
  }

  const int rowh = row_base + ((lane >> 4) << 3);
#pragma unroll
  for (int t = 0; t < NT; ++t) {
    const int col = t * 16 + r;
#pragma unroll
    for (int j = 0; j < 8; ++j)
      H[(size_t)(rowh + j) * NC + col] = acc[t][j];
  }
}

// ---------------- aggregation ----------------

// agg[i,d] = dinv[i]^2 * h[i,d] + bias[d]   (self-loop message + bias; atomics add on top)
__global__ void init_agg_kernel(const float* __restrict__ h, const float* __restrict__ dinv,
                                const float* __restrict__ bias, float* __restrict__ agg,
                                int n, int D) {
  int idx = blockIdx.x * blockDim.x + threadIdx.x;
  if (idx >= n * D) return;
  int i = idx / D;
  int d = idx - i * D;
  float di = dinv[i];
  agg[idx] = di * di * h[idx] + bias[d];
}

// one wave per edge, 4 floats per lane (D=128)
__global__ __launch_bounds__(256) void scatter_d128(const int* __restrict__ src,
                                                    const int* __restrict__ tgt,
                                                    const float* __restrict__ dinv,
                                                    const float* __restrict__ h,
                                                    float* __restrict__ agg, int n_edges) {
  int e = blockIdx.x * 8 + (threadIdx.x >> 5);
  if (e >= n_edges) return;
  int lane = threadIdx.x & 31;
  int s = src[e], t = tgt[e];
  float w = dinv[s] * dinv[t];
  const float4 v = *(const float4*)(h + (size_t)s * 128 + lane * 4);
  float* ap = agg + (size_t)t * 128 + lane * 4;
  atomicAdd(ap + 0, w * v.x);
  atomicAdd(ap + 1, w * v.y);
  atomicAdd(ap + 2, w * v.z);
  atomicAdd(ap + 3, w * v.w);
}

// one wave per edge, 2 floats per lane (D=64)
__global__ __launch_bounds__(256) void scatter_d64(const int* __restrict__ src,
                                                   const int* __restrict__ tgt,
                                                   const float* __restrict__ dinv,
                                                   const float* __restrict__ h,
                                                   float* __restrict__ agg, int n_edges) {
  int e = blockIdx.x * 8 + (threadIdx.x >> 5);
  if (e >= n_edges) return;
  int lane = threadIdx.x & 31;
  int s = src[e], t = tgt[e];
  float w = dinv[s] * dinv[t];
  const float2 v = *(const float2*)(h + (size_t)s * 64 + lane * 2);
  float* ap = agg + (size_t)t * 64 + lane * 2;
  atomicAdd(ap + 0, w * v.x);
  atomicAdd(ap + 1, w * v.y);
}

__global__ void relu_kernel(float* __restrict__ a, int n) {
  int i = blockIdx.x * blockDim.x + threadIdx.x;
  if (i < n) a[i] = fmaxf(a[i], 0.0f);
}

// ---------------- launch ----------------

static inline size_t align_up(size_t x, size_t a) { return (x + a - 1) & ~(a - 1); }

extern "C" void kernel_launch(void* const* d_in, const int* in_sizes, int n_in,
                              void* d_out, int out_size, void* d_ws, size_t ws_size,
                              hipStream_t stream) {
  const float* x  = (const float*)d_in[0];
  const int*   ei = (const int*)d_in[1];
  const float* W1 = (const float*)d_in[2];
  const float* b1 = (const float*)d_in[3];
  const float* W2 = (const float*)d_in[4];
  const float* b2 = (const float*)d_in[5];

  const int N = in_sizes[0] / IN_DIM;   // 100000
  const int E = in_sizes[1] / 2;        // 1600000
  const int* src = ei;                  // edge_index[0]
  const int* tgt = ei + E;              // edge_index[1]

  // workspace layout (H2 aliases H1 — H1 is dead after layer-1 scatter)
  char* ws = (char*)d_ws;
  size_t off = 0;
  float* dinv = (float*)(ws + off); off = align_up(off + (size_t)N * 4, 256);
  float* H1   = (float*)(ws + off); off = align_up(off + (size_t)N * IN_DIM * 4, 256);
  float* A1   = (float*)(ws + off); off = align_up(off + (size_t)N * IN_DIM * 4, 256);
  float* H2   = H1;
  float* out  = (float*)d_out;

  const int TB = 256;
  dim3 blk(TB);

  // normalization
  fill_deg_kernel<<<dim3((N + TB - 1) / TB), blk, 0, stream>>>(dinv, N);
  deg_accum_kernel<<<dim3((E + TB - 1) / TB), blk, 0, stream>>>(tgt, dinv, E);
  rsqrt_kernel<<<dim3((N + TB - 1) / TB), blk, 0, stream>>>(dinv, N);

  const int gemm_blocks = (N + 127) / 128;
  const int eblocks = (E + 7) / 8;

  // layer 1
  gemm_wmma_f32<128><<<dim3(gemm_blocks), blk, 0, stream>>>(x, W1, H1, N);
  init_agg_kernel<<<dim3((N * 128 + TB - 1) / TB), blk, 0, stream>>>(H1, dinv, b1, A1, N, 128);
  scatter_d128<<<dim3(eblocks), blk, 0, stream>>>(src, tgt, dinv, H1, A1, E);
  relu_kernel<<<dim3((N * 128 + TB - 1) / TB), blk, 0, stream>>>(A1, N * 128);

  // layer 2
  gemm_wmma_f32<64><<<dim3(gemm_blocks), blk, 0, stream>>>(A1, W2, H2, N);
  init_agg_kernel<<<dim3((N * 64 + TB - 1) / TB), blk, 0, stream>>>(H2, dinv, b2, out, N, 64);
  scatter_d64<<<dim3(eblocks), blk, 0, stream>>>(src, tgt, dinv, H2, out, E);
}